// PhysicsEngine_32856499815037
// MI455X (gfx1250) — compile-verified
//
#include <hip/hip_runtime.h>
#include <hip/hip_bf16.h>
#include <float.h>

// Problem dimensions (from reference)
#define BB 16
#define LL 128
#define PP 8192
#define JSPLIT 32                         // j-range chunks per batch
#define JTILES (PP / 16)                  // 512 column tiles per batch
#define TILES_PER_CHUNK (JTILES / JSPLIT) // 16 tiles (256 columns) per block

typedef __attribute__((ext_vector_type(2))) float v2f;
typedef __attribute__((ext_vector_type(8))) float v8f;

// --------------------------- fast math (native CDNA5 trans ops) -------------
__device__ __forceinline__ float frcp(float x)  { return __builtin_amdgcn_rcpf(x);  }  // v_rcp_f32
__device__ __forceinline__ float fsqrt(float x) { return __builtin_amdgcn_sqrtf(x); }  // v_sqrt_f32
__device__ __forceinline__ float frsq(float x)  { return __builtin_amdgcn_rsqf(x);  }  // v_rsq_f32
__device__ __forceinline__ float fexp2(float x) { return __builtin_amdgcn_exp2f(x); }  // v_exp_f32

__device__ __forceinline__ float ftanh(float x) {
#if __has_builtin(__builtin_amdgcn_tanhf)
  return __builtin_amdgcn_tanhf(x);                       // v_tanh_f32 (gfx1250)
#else
  // tanh(x) = 1 - 2/(1 + e^{2x});  e^{2x} = 2^{2x*log2(e)}   (branch-free)
  float e = fexp2(x * 2.8853900817779268f);
  return 1.0f - 2.0f * frcp(1.0f + e);
#endif
}
__device__ __forceinline__ float fsigmoid(float x) {      // 1/(1+e^{-x})
  return frcp(1.0f + fexp2(x * -1.4426950408889634f));
}

// ---------------------------------------------------------------------------
// Ligand precompute: radii, relu(eps), |pos|^2
// ---------------------------------------------------------------------------
__global__ void precomp_L(const float* __restrict__ pos_L,
                          const float* __restrict__ x_L,
                          const float* __restrict__ vdw_radii,
                          const float* __restrict__ epsilon_tbl,
                          float* __restrict__ radL,
                          float* __restrict__ epsL,
                          float* __restrict__ lsq) {
  int i = blockIdx.x * blockDim.x + threadIdx.x;
  if (i >= BB * LL) return;
  const float* x = x_L + (size_t)i * 9;
  float r = 0.0f, e = 0.0f;
#pragma unroll
  for (int k = 0; k < 9; ++k) {
    float xv = x[k];
    r += xv * vdw_radii[k];
    e += xv * epsilon_tbl[k];
  }
  radL[i] = r;
  epsL[i] = fmaxf(e, 0.0f);     // relu before pairwise product
  const float* p = pos_L + (size_t)i * 3;
  lsq[i] = p[0] * p[0] + p[1] * p[1] + p[2] * p[2];
}

// ---------------------------------------------------------------------------
// Protein precompute, packed per column: {px,py,pz,|p|^2} {qP,radP,epsP,xP0}
// ---------------------------------------------------------------------------
__global__ void precomp_P(const float* __restrict__ pos_P,
                          const float* __restrict__ x_P,
                          const float* __restrict__ q_P,
                          float4* __restrict__ pkP) {   // 2 float4 per column
  int i = blockIdx.x * blockDim.x + threadIdx.x;
  if (i >= BB * PP) return;
  const float* x = x_P + (size_t)i * 4;
  const float* p = pos_P + (size_t)i * 3;
  float4 c0, c1;
  c0.x = p[0]; c0.y = p[1]; c0.z = p[2];
  c0.w = p[0] * p[0] + p[1] * p[1] + p[2] * p[2];
  c1.x = q_P[i];
  c1.y = x[0] * 1.70f + x[1] * 1.55f + x[2] * 1.52f + x[3] * 1.80f;  // radP
  c1.z = x[0] * 0.10f + x[1] * 0.10f + x[2] * 0.15f + x[3] * 0.20f;  // epsP
  c1.w = x[0];                                                        // xP0
  pkP[(size_t)i * 2 + 0] = c0;
  pkP[(size_t)i * 2 + 1] = c1;
}

// ---------------------------------------------------------------------------
// Main pairwise kernel. Wave w owns i-tile w (16 ligand rows); the block's
// 256 protein columns are staged in LDS once. dist_sq tiles come from
// V_WMMA_F32_16X16X4_F32:  D = lsq (C) + A[lx,ly,lz,1] . B[-2px,-2py,-2pz,psq]
// ---------------------------------------------------------------------------
__global__ void __launch_bounds__(256)
pair_kernel(const float* __restrict__ pos_L,
            const float* __restrict__ q_L,
            const float* __restrict__ x_L,
            const float* __restrict__ radL,  const float* __restrict__ epsL,
            const float* __restrict__ lsq,
            const float4* __restrict__ pkP,
            float* __restrict__ partials) {
  __shared__ float4 sP[256 * 2];   // 8 KB: this block's 256 packed columns
  __shared__ float red[8][4];

  const int tid  = threadIdx.x;
  const int w    = tid >> 5;        // wave id == i-tile id (0..7), covers L=128
  const int lane = tid & 31;
  const int half = lane >> 4;       // 0: rows 0-7 / K=0,1 ; 1: rows 8-15 / K=2,3
  const int col  = lane & 15;       // N index of this lane
  const int b     = blockIdx.x / JSPLIT;
  const int chunk = blockIdx.x % JSPLIT;
  const int i0    = w * 16;

  // ---- stage 256 packed protein columns into LDS (one column per thread)
  {
    size_t pi = (size_t)b * PP + (size_t)chunk * 256 + tid;
    sP[tid * 2 + 0] = pkP[pi * 2 + 0];
    sP[tid * 2 + 1] = pkP[pi * 2 + 1];
  }

  // ---- loop-invariant per-row ligand data (C/D VGPR r holds M = r + 8*half)
  float qLr[8], radLr[8], epsLr[8], xL0r[8];
  v8f cfrag;
#pragma unroll
  for (int r = 0; r < 8; ++r) {
    int m = i0 + r + 8 * half;
    size_t li = (size_t)b * LL + m;
    qLr[r]   = q_L[li];
    radLr[r] = radL[li];
    epsLr[r] = epsL[li];
    xL0r[r]  = x_L[li * 9];   // type_probs_L[..., 0]
    cfrag[r] = lsq[li];       // |l|^2 folded into the accumulator
  }

  // ---- A fragment: row M = lane%16; lanes<16 hold K=0,1 ; lanes>=16 K=2,3
  size_t ai = ((size_t)b * LL + i0 + col) * 3;
  float lx = pos_L[ai + 0], ly = pos_L[ai + 1], lz = pos_L[ai + 2];
  v2f afrag;
  afrag.x = half ? lz : lx;       // K=2 : K=0
  afrag.y = half ? 1.0f : ly;     // K=3 : K=1

  __syncthreads();

  float s1 = 0.0f, s2 = 0.0f, s3 = 0.0f, s4 = 0.0f;

  for (int jt = 0; jt < TILES_PER_CHUNK; ++jt) {
    int ci = jt * 16 + col;
    float4 c0 = sP[ci * 2 + 0];   // px,py,pz,|p|^2
    float4 c1 = sP[ci * 2 + 1];   // qP,radP,epsP,xP0

    // B fragment: col N = lane%16; lanes<16 hold K=0,1 ; lanes>=16 hold K=2,3
    v2f bfrag;
    bfrag.x = half ? (-2.0f * c0.z) : (-2.0f * c0.x);
    bfrag.y = half ? c0.w           : (-2.0f * c0.y);

    // D[r][lane] = dist_sq(M = r + 8*half, N = col)
    v8f d = __builtin_amdgcn_wmma_f32_16x16x4_f32(
        /*neg_a=*/false, afrag, /*neg_b=*/false, bfrag,
        /*c_mod=*/(short)0, cfrag, /*reuse_a=*/false, /*reuse_b=*/false);

#pragma unroll
    for (int r = 0; r < 8; ++r) {
      float dsq  = fmaxf(d[r], 0.0f);                 // guard cancellation
      float dist = fsqrt(dsq + 1e-8f);
      float sigma = radLr[r] + c1.y;
      // both uses of soft_dist are reciprocal uses -> one v_rsq_f32
      float inv_soft = frsq(dsq + 2.0f * sigma * sigma + 1e-8f);   // ALPHA=2
      // 332.06/4 = 83.015 ; then /200 inside tanh
      float e_elec = 200.0f * ftanh(83.015f * qLr[r] * c1.x * inv_soft * 0.005f);
      float eps_ij = fsqrt(epsLr[r] * c1.z + 1e-8f);
      float ratio  = fminf(sigma * inv_soft, 5.0f);
      float r2 = ratio * ratio;
      float r6 = r2 * r2 * r2;
      float evdw = 4.0f * eps_ij * (r6 * r6 - r6);
      float epos = 500.0f * ftanh(fmaxf(evdw, 0.0f) * 0.002f);
      float eneg = -10.0f * ftanh(fmaxf(-evdw, 0.0f) * 0.1f);
      float dm = fsigmoid(2.0f * (12.0f - dist));

      s1 += (e_elec + epos + eneg) * dm;              // e_soft core
      float t = dist * 0.25f;
      float t2 = t * t;
      float hsa = frcp(1.0f + t2 * t2);
      s2 += -0.5f * (xL0r[r] * c1.w) * hsa * dm;      // e_hsa
      float ov = fmaxf(sigma * 0.6f - dist, 0.0f);
      s3 += ov * ov;                                  // pauli overlap^2
      float lvdw = fminf(fmaxf(evdw, -10.0f), 500.0f);
      s4 += (e_elec + lvdw) * dm;                     // log_soft core
    }
  }

  // ---- wave32 butterfly reduction (deterministic)
#pragma unroll
  for (int o = 16; o > 0; o >>= 1) {
    s1 += __shfl_xor(s1, o, 32);
    s2 += __shfl_xor(s2, o, 32);
    s3 += __shfl_xor(s3, o, 32);
    s4 += __shfl_xor(s4, o, 32);
  }
  if (lane == 0) {
    red[w][0] = s1; red[w][1] = s2; red[w][2] = s3; red[w][3] = s4;
  }
  __syncthreads();
  if (tid == 0) {
    float t1 = 0.0f, t2 = 0.0f, t3 = 0.0f, t4 = 0.0f;
    for (int k = 0; k < 8; ++k) {   // fixed order -> deterministic
      t1 += red[k][0]; t2 += red[k][1]; t3 += red[k][2]; t4 += red[k][3];
    }
    size_t base = (size_t)blockIdx.x * 4;
    partials[base + 0] = t1;
    partials[base + 1] = t2;
    partials[base + 2] = t3;
    partials[base + 3] = t4;
  }
}

// ---------------------------------------------------------------------------
// Final scalar math + packing: e_raw[16] | e_hard_log[16] | alpha | log[16]
// ---------------------------------------------------------------------------
__global__ void finalize_kernel(const float* __restrict__ partials,
                                float* __restrict__ out) {
  int bidx = threadIdx.x;
  if (bidx == 0) out[32] = 2.0f;  // ALPHA
  if (bidx >= BB) return;
  float s1 = 0.0f, s2 = 0.0f, s3 = 0.0f, s4 = 0.0f;
  for (int c = 0; c < JSPLIT; ++c) {  // fixed order -> deterministic
    size_t base = ((size_t)(bidx * JSPLIT + c)) * 4;
    s1 += partials[base + 0];
    s2 += partials[base + 1];
    s3 += partials[base + 2];
    s4 += partials[base + 3];
  }
  const float tgate = fsigmoid(3.0f);   // sigmoid((0.5-0.2)*10)
  float e_hsa   = s2;
  float e_pauli = tgate * 100.0f * s3;
  float e_raw   = s1 + 5.0f * e_hsa + e_pauli;
  if (isnan(e_raw))      e_raw = 0.0f;                       // nan_to_num
  else if (isinf(e_raw)) e_raw = (e_raw > 0.0f) ? FLT_MAX : -FLT_MAX;
  float e_hard_log = fminf(e_pauli, 10000.0f);
  float e_soft_log = fminf(fmaxf(s4 + 5.0f * e_hsa, -500.0f), 5000.0f);
  float log_energy = fminf(e_soft_log + e_hard_log, 1000000.0f);
  out[bidx]      = e_raw;
  out[16 + bidx] = e_hard_log;
  out[33 + bidx] = log_energy;
}

// ---------------------------------------------------------------------------
extern "C" void kernel_launch(void* const* d_in, const int* in_sizes, int n_in,
                              void* d_out, int out_size, void* d_ws, size_t ws_size,
                              hipStream_t stream) {
  const float* pos_L = (const float*)d_in[0];
  const float* pos_P = (const float*)d_in[1];
  const float* q_L   = (const float*)d_in[2];
  const float* q_P   = (const float*)d_in[3];
  const float* x_L   = (const float*)d_in[4];
  const float* x_P   = (const float*)d_in[5];
  const float* vdw   = (const float*)d_in[6];
  const float* epst  = (const float*)d_in[7];
  float* out = (float*)d_out;
  float* ws  = (float*)d_ws;

  // Workspace layout (floats): L-side arrays + packed P (~4.2 MB) + partials
  float*  radL = ws;                               // BB*LL
  float*  epsL = radL + (size_t)BB * LL;           // BB*LL
  float*  lsq  = epsL + (size_t)BB * LL;           // BB*LL
  float4* pkP  = (float4*)(lsq + (size_t)BB * LL); // BB*PP * 2 float4
  float*  partials = (float*)(pkP + (size_t)BB * PP * 2);  // (BB*JSPLIT)*4

  precomp_L<<<(BB * LL + 255) / 256, 256, 0, stream>>>(pos_L, x_L, vdw, epst,
                                                       radL, epsL, lsq);
  precomp_P<<<(BB * PP + 255) / 256, 256, 0, stream>>>(pos_P, x_P, q_P, pkP);
  pair_kernel<<<BB * JSPLIT, 256, 0, stream>>>(pos_L, q_L, x_L,
                                               radL, epsL, lsq, pkP, partials);
  finalize_kernel<<<1, 32, 0, stream>>>(partials, out);
}